// MultiHeadGraphAttention_23854248362786
// MI455X (gfx1250) — compile-verified
//
#include <hip/hip_runtime.h>
#include <hip/hip_bf16.h>

typedef __attribute__((ext_vector_type(2))) float v2f;
typedef __attribute__((ext_vector_type(4))) float v4f;
typedef __attribute__((ext_vector_type(8))) float v8f;

#define N_TOK    512
#define DMODEL   512
#define NHEADS   8
#define DKH      64
#define BATCH    2
#define BN       (BATCH * N_TOK)        /* 1024 rows in (B*N, D) matrices */
#define MAT_ELTS (BN * DMODEL)          /* 524288 floats = 2MB            */

// ---------------- wave32 reductions ----------------
static __device__ __forceinline__ float wave_sum(float v) {
#pragma unroll
  for (int m = 16; m >= 1; m >>= 1) v += __shfl_xor(v, m, 32);
  return v;
}
static __device__ __forceinline__ float wave_max(float v) {
#pragma unroll
  for (int m = 16; m >= 1; m >>= 1) v = fmaxf(v, __shfl_xor(v, m, 32));
  return v;
}

// ---------------- f32 WMMA 16x16x4 ----------------
static __device__ __forceinline__ v8f wmma4(v2f a, v2f b, v8f c) {
  // 8 args: (neg_a, A, neg_b, B, c_mod, C, reuse_a, reuse_b)
  return __builtin_amdgcn_wmma_f32_16x16x4_f32(false, a, false, b, (short)0, c,
                                               false, false);
}

// ============================================================
// K1: edge bias = mean over last dim of edge_features (B,N,N,D)
// one wave per (b,i,j) row of 512 floats. 1.07GB streamed exactly
// once -> non-temporal loads so the QKV/attn working set stays in L2.
// ============================================================
__global__ void edge_bias_kernel(const float* __restrict__ edge,
                                 float* __restrict__ bias) {
  int row  = (blockIdx.x * blockDim.x + threadIdx.x) >> 5;
  int lane = threadIdx.x & 31;
  const float* p = edge + (size_t)row * DMODEL;
  float s = 0.f;
#pragma unroll
  for (int k = 0; k < 4; ++k) {
    v4f v = __builtin_nontemporal_load((const v4f*)(p + lane * 4 + k * 128));
    s += (v.x + v.y) + (v.z + v.w);
  }
  s = wave_sum(s);
  if (lane == 0) bias[row] = s * (1.0f / 512.0f);
}

// ============================================================
// K2/K6: Out = X @ W^T + bvec (+ resid), X:(1024,512) W:(512,512)
// one wave per 16x16 C tile, K stepped by 4 with f32 WMMA.
// RESID is a compile-time flag -> branch-free epilogue.
// ============================================================
template <bool RESID>
__global__ void gemm_xwt_kernel(const float* __restrict__ X,
                                const float* __restrict__ W,
                                const float* __restrict__ bvec,
                                const float* __restrict__ resid,
                                float* __restrict__ Out) {
  int wave = threadIdx.x >> 5, lane = threadIdx.x & 31;
  int t  = blockIdx.x * 8 + wave;          // 0..2047 tiles (64 x 32)
  int m0 = (t >> 5) << 4;
  int n0 = (t & 31) << 4;
  int r  = lane & 15, hi = lane >> 4, kb = hi * 2;
  const float* Arow = X + (size_t)(m0 + r) * DMODEL;   // A[m][k]
  const float* Brow = W + (size_t)(n0 + r) * DMODEL;   // B[k][n] = W[n][k]
  v8f acc = {};
#pragma unroll 4
  for (int k0 = 0; k0 < DMODEL; k0 += 4) {
    float2 af = *(const float2*)(Arow + k0 + kb);
    float2 bf = *(const float2*)(Brow + k0 + kb);
    acc = wmma4((v2f){af.x, af.y}, (v2f){bf.x, bf.y}, acc);
  }
  int col  = n0 + r;
  float bv = bvec[col];
#pragma unroll
  for (int v = 0; v < 8; ++v) {
    int m = m0 + v + hi * 8;
    float val = acc[v] + bv;
    if (RESID) val += resid[(size_t)m * DMODEL + col];
    Out[(size_t)m * DMODEL + col] = val;
  }
}

// ============================================================
// K3: scores = Qh Kh^T / 8 + edge_bias, masked; -> attn (raw)
// grid: (128, 16=b*8+h), one wave per 16x16 tile of (512,512)
// ============================================================
__global__ void scores_kernel(const float* __restrict__ Qp,
                              const float* __restrict__ Kp,
                              const float* __restrict__ ebias,
                              const int* __restrict__ mask,
                              float* __restrict__ attn) {
  int wave = threadIdx.x >> 5, lane = threadIdx.x & 31;
  int bh = blockIdx.y, b = bh >> 3, h = bh & 7;
  int t  = blockIdx.x * 8 + wave;          // 0..1023 tiles (32 x 32)
  int m0 = (t >> 5) << 4;
  int n0 = (t & 31) << 4;
  int r  = lane & 15, hi = lane >> 4, kb = hi * 2;
  const float* Arow = Qp + (size_t)(b * N_TOK + m0 + r) * DMODEL + h * DKH;
  const float* Brow = Kp + (size_t)(b * N_TOK + n0 + r) * DMODEL + h * DKH;
  v8f acc = {};
#pragma unroll
  for (int k0 = 0; k0 < DKH; k0 += 4) {
    float2 af = *(const float2*)(Arow + k0 + kb);
    float2 bf = *(const float2*)(Brow + k0 + kb);
    acc = wmma4((v2f){af.x, af.y}, (v2f){bf.x, bf.y}, acc);
  }
  int j = n0 + r;
#pragma unroll
  for (int v = 0; v < 8; ++v) {
    int i = m0 + v + hi * 8;
    size_t bij = (size_t)(b * N_TOK + i) * N_TOK + j;
    float sc = acc[v] * 0.125f + ebias[bij];
    sc = (mask[bij] == 0) ? -1e9f : sc;
    attn[((size_t)bh * N_TOK + i) * N_TOK + j] = sc;
  }
}

// ============================================================
// K4: row softmax in-place on attn (B*H*N rows of 512)
// ============================================================
__global__ void softmax_kernel(float* __restrict__ attn) {
  int row  = (blockIdx.x * blockDim.x + threadIdx.x) >> 5;
  int lane = threadIdx.x & 31;
  float* p = attn + (size_t)row * N_TOK;
  float4 v[4];
  float mx = -3.4e38f;
#pragma unroll
  for (int k = 0; k < 4; ++k) {
    v[k] = *(const float4*)(p + lane * 4 + k * 128);
    mx = fmaxf(mx, fmaxf(fmaxf(v[k].x, v[k].y), fmaxf(v[k].z, v[k].w)));
  }
  mx = wave_max(mx);
  float s = 0.f;
#pragma unroll
  for (int k = 0; k < 4; ++k) {
    v[k].x = __expf(v[k].x - mx);
    v[k].y = __expf(v[k].y - mx);
    v[k].z = __expf(v[k].z - mx);
    v[k].w = __expf(v[k].w - mx);
    s += (v[k].x + v[k].y) + (v[k].z + v[k].w);
  }
  s = wave_sum(s);
  float inv = 1.0f / s;
#pragma unroll
  for (int k = 0; k < 4; ++k) {
    float4 o = {v[k].x * inv, v[k].y * inv, v[k].z * inv, v[k].w * inv};
    *(float4*)(p + lane * 4 + k * 128) = o;
  }
}

// ============================================================
// K5: ctx[b,i,h*64+dk] = attn[b,h,i,:] @ V[b,:,h*64+dk]
// one wave per 16x16 tile, per (b,h): 32x4 tiles, K=512
// ============================================================
__global__ void ctx_kernel(const float* __restrict__ attn,
                           const float* __restrict__ Vp,
                           float* __restrict__ ctx) {
  int wave = threadIdx.x >> 5, lane = threadIdx.x & 31;
  int t  = blockIdx.x * 8 + wave;          // 0..2047
  int bh = t >> 7, tt = t & 127;
  int b  = bh >> 3, h = bh & 7;
  int m0 = (tt >> 2) << 4;
  int n0 = (tt & 3) << 4;
  int r  = lane & 15, hi = lane >> 4, kb = hi * 2;
  const float* Arow = attn + ((size_t)bh * N_TOK + m0 + r) * N_TOK;
  const float* Bcol = Vp + (size_t)(b * N_TOK) * DMODEL + h * DKH + n0 + r;
  v8f acc = {};
#pragma unroll 4
  for (int k0 = 0; k0 < N_TOK; k0 += 4) {
    float2 af = *(const float2*)(Arow + k0 + kb);
    float bx = Bcol[(size_t)(k0 + kb) * DMODEL];
    float by = Bcol[(size_t)(k0 + kb + 1) * DMODEL];
    acc = wmma4((v2f){af.x, af.y}, (v2f){bx, by}, acc);
  }
  int col = h * DKH + n0 + r;
#pragma unroll
  for (int v = 0; v < 8; ++v) {
    int m = m0 + v + hi * 8;
    ctx[(size_t)(b * N_TOK + m) * DMODEL + col] = acc[v];
  }
}

// ============================================================
// K7: LayerNorm over rows of x (1024 x 512), biased variance
// ============================================================
__global__ void layernorm_kernel(const float* __restrict__ x,
                                 const float* __restrict__ lw,
                                 const float* __restrict__ lb,
                                 float* __restrict__ out) {
  int row  = (blockIdx.x * blockDim.x + threadIdx.x) >> 5;
  int lane = threadIdx.x & 31;
  const float* p = x + (size_t)row * DMODEL;
  float4 v[4];
  float s = 0.f, ss = 0.f;
#pragma unroll
  for (int k = 0; k < 4; ++k) {
    v[k] = *(const float4*)(p + lane * 4 + k * 128);
    s  += (v[k].x + v[k].y) + (v[k].z + v[k].w);
    ss += (v[k].x * v[k].x + v[k].y * v[k].y) + (v[k].z * v[k].z + v[k].w * v[k].w);
  }
  s  = wave_sum(s);
  ss = wave_sum(ss);
  float mu   = s * (1.0f / 512.0f);
  float var  = ss * (1.0f / 512.0f) - mu * mu;
  float rstd = rsqrtf(var + 1e-5f);
#pragma unroll
  for (int k = 0; k < 4; ++k) {
    int j = lane * 4 + k * 128;
    float4 wv = *(const float4*)(lw + j);
    float4 bv = *(const float4*)(lb + j);
    float4 o;
    o.x = (v[k].x - mu) * rstd * wv.x + bv.x;
    o.y = (v[k].y - mu) * rstd * wv.y + bv.y;
    o.z = (v[k].z - mu) * rstd * wv.z + bv.z;
    o.w = (v[k].w - mu) * rstd * wv.w + bv.w;
    *(float4*)(out + (size_t)row * DMODEL + j) = o;
  }
}

// ============================================================
extern "C" void kernel_launch(void* const* d_in, const int* in_sizes, int n_in,
                              void* d_out, int out_size, void* d_ws,
                              size_t ws_size, hipStream_t stream) {
  const float* query = (const float*)d_in[0];
  const float* key_  = (const float*)d_in[1];
  const float* value = (const float*)d_in[2];
  const float* edge  = (const float*)d_in[3];
  const int*   mask  = (const int*)d_in[4];
  const float* Wq = (const float*)d_in[5];
  const float* bq = (const float*)d_in[6];
  const float* Wk = (const float*)d_in[7];
  const float* bk = (const float*)d_in[8];
  const float* Wv = (const float*)d_in[9];
  const float* bv = (const float*)d_in[10];
  const float* Wo = (const float*)d_in[11];
  const float* bo = (const float*)d_in[12];
  const float* lw = (const float*)d_in[13];
  const float* lb = (const float*)d_in[14];

  float* out_main = (float*)d_out;                 // (B,N,D)  524288
  float* attn     = (float*)d_out + MAT_ELTS;      // (B,H,N,N) 4194304

  float* ws   = (float*)d_ws;
  float* Qp   = ws;                 // (1024,512)
  float* Kp   = ws + 1 * MAT_ELTS;
  float* Vp   = ws + 2 * MAT_ELTS;
  float* ebia = ws + 3 * MAT_ELTS;  // (B,N,N)
  float* ctx  = Qp;                 // reuse: Qp dead after scores
  float* xbuf = Kp;                 // reuse: Kp dead after scores

  const int TB = 256;  // 8 waves

  // K1: edge bias, 524288 rows / 8 waves = 65536 blocks (1GB NT sweep)
  edge_bias_kernel<<<dim3(65536), TB, 0, stream>>>(edge, ebia);

  // K2: Q/K/V projections (2048 tiles each -> 256 blocks)
  gemm_xwt_kernel<false><<<dim3(256), TB, 0, stream>>>(query, Wq, bq, nullptr, Qp);
  gemm_xwt_kernel<false><<<dim3(256), TB, 0, stream>>>(key_,  Wk, bk, nullptr, Kp);
  gemm_xwt_kernel<false><<<dim3(256), TB, 0, stream>>>(value, Wv, bv, nullptr, Vp);

  // K3: raw scores -> attn region of d_out
  scores_kernel<<<dim3(128, 16), TB, 0, stream>>>(Qp, Kp, ebia, mask, attn);

  // K4: softmax in place (8192 rows / 8 waves)
  softmax_kernel<<<dim3(1024), TB, 0, stream>>>(attn);

  // K5: ctx = attn @ V  (2048 tiles)
  ctx_kernel<<<dim3(256), TB, 0, stream>>>(attn, Vp, ctx);

  // K6: output projection + residual (query)
  gemm_xwt_kernel<true><<<dim3(256), TB, 0, stream>>>(ctx, Wo, bo, query, xbuf);

  // K7: LayerNorm -> main output (1024 rows / 8 waves)
  layernorm_kernel<<<dim3(128), TB, 0, stream>>>(xbuf, lw, lb, out_main);
}